// SSD_83176336654490
// MI455X (gfx1250) — compile-verified
//
#include <hip/hip_runtime.h>
#include <hip/hip_bf16.h>

typedef __attribute__((ext_vector_type(2))) float v2f;
typedef __attribute__((ext_vector_type(8))) float v8f;
typedef unsigned int u32;
typedef __attribute__((ext_vector_type(4))) u32 u32x4;
typedef __attribute__((ext_vector_type(4))) int  i32x4;
typedef __attribute__((ext_vector_type(8))) int  i32x8;

#define BATCH 16
#define DM    1024
#define DM2   512
#define LTOT  4096
#define SDIM  64

__device__ __forceinline__ v8f wmma4(v2f a, v2f b, v8f c) {
    // V_WMMA_F32_16X16X4_F32 : D(16x16 f32) = A(16x4 f32) * B(4x16 f32) + C
    return __builtin_amdgcn_wmma_f32_16x16x4_f32(
        /*neg_a=*/false, a, /*neg_b=*/false, b,
        /*c_mod=*/(short)0, c, /*reuse_a=*/false, /*reuse_b=*/false);
}

// ---------------------------------------------------------------------------
// TDM: load a 64(s) x 64(d) f32 tile of w_bcdt (row-major 64x512) into LDS.
// D# per CDNA5 ISA ch.8: group0 = {count=1 | lds_addr | global_addr | type=2},
// group1 = {data_size=4B, tensor_dim0=512, tensor_dim1=64, tile_dim0=64,
//           tile_dim1=64, tensor_dim0_stride=512}, groups 2/3 = 0 (2-D tensor).
// All inputs are block-uniform -> SGPR descriptor groups.
// ---------------------------------------------------------------------------
__device__ __forceinline__ void tdm_load_w_tile(u32 lds_byte_off, const float* gsrc) {
    unsigned long long ga = (unsigned long long)gsrc;
    u32x4 g0;
    g0.x = 1u;                                            // count=1, user D#
    g0.y = lds_byte_off;                                  // lds_addr (bytes)
    g0.z = (u32)(ga & 0xffffffffull);                     // global_addr[31:0]
    g0.w = (u32)((ga >> 32) & 0x1ffffffull) | (2u << 30); // ga[56:32] | type=2
    i32x8 g1 = { 0x20000,          // data_size=2 (4 bytes); mask/flags = 0
                 0x02000000,       // tensor_dim0[15:0]=512 @ bits63:48
                 (64 << 16),       // tensor_dim1[15:0]=64  @ bits95:80
                 (64 << 16),       // tile_dim0=64          @ bits127:112
                 64,               // tile_dim1=64          @ bits143:128
                 512,              // tensor_dim0_stride[31:0]=512 @ bits191:160
                 0, 0 };
    i32x4 gz = {0, 0, 0, 0};
#if __has_include(<hip/amd_detail/amd_gfx1250_TDM.h>)
    // therock-10.0 headers present -> 6-arg builtin (extra i32x8 group)
    i32x8 gz8 = {0, 0, 0, 0, 0, 0, 0, 0};
    __builtin_amdgcn_tensor_load_to_lds(g0, g1, gz, gz, gz8, 0);
#else
    // ROCm 7.2 clang-22 -> 5-arg builtin
    __builtin_amdgcn_tensor_load_to_lds(g0, g1, gz, gz, 0);
#endif
}

// -------------------------------------------------------------------------
// Kernel 1:  BCdt_raw[b,s,l] = sum_d w_bcdt[s,d]*xh[b,d,l] + b_bcdt[s]
// Grid (32 l-slabs, 16 b), 256 thr = 8 waves; wave -> 16 l cols x all 64 s.
// A operand (w_bcdt) staged through LDS by the Tensor Data Mover, double
// buffered in 64-d chunks; B operand (xh) streamed from HBM.
// -------------------------------------------------------------------------
__global__ __launch_bounds__(256) void k_bcdt(const float* __restrict__ x,
                                              const float* __restrict__ w_bcdt,
                                              const float* __restrict__ b_bcdt,
                                              float* __restrict__ raw) {
    __shared__ float sw[2][SDIM * 64];   // 2 x 16 KB double buffer
    const int lane = threadIdx.x & 31;
    const int wave = threadIdx.x >> 5;
    const int ln = lane & 15, hi = lane >> 4;
    const int b = blockIdx.y;
    const int l = blockIdx.x * 128 + wave * 16 + ln;
    const float* __restrict__ xh = x + (size_t)b * DM * LTOT;

    v8f zero = {0.f,0.f,0.f,0.f,0.f,0.f,0.f,0.f};
    v8f acc[4] = {zero, zero, zero, zero};

    // kick chunk 0 (d = 0..63) via TDM
    if (wave == 0)
        tdm_load_w_tile((u32)(unsigned long long)&sw[0][0], w_bcdt);

    for (int kc = 0; kc < 8; ++kc) {             // 8 chunks of 64 d
        if (wave == 0)
            __builtin_amdgcn_s_wait_tensorcnt(0);   // current chunk landed in LDS
        __syncthreads();
        if (wave == 0 && kc + 1 < 8)                // overlap: fetch next chunk
            tdm_load_w_tile((u32)(unsigned long long)&sw[(kc + 1) & 1][0],
                            w_bcdt + (kc + 1) * 64);

        const float* wbuf = sw[kc & 1];
        const int kbase = kc * 64;
        for (int kk = 0; kk < 64; kk += 4) {
            const int k = kbase + kk;
            // B operand (4x16 f32): lanes 0-15 K={k,k+1}, lanes 16-31 K={k+2,k+3}
            v2f bb;
            bb.x = xh[(size_t)(k + 2*hi)     * LTOT + l];
            bb.y = xh[(size_t)(k + 2*hi + 1) * LTOT + l];
            if (kk == 0 && kc + 1 < 8)
                __builtin_prefetch(&xh[(size_t)(kbase + 64 + 2*hi) * LTOT + l], 0, 3);
#pragma unroll
            for (int st = 0; st < 4; ++st) {
                // A operand from LDS chunk: row s = st*16+ln, cols kk..kk+3
                v2f aa = *(const v2f*)(wbuf + (st*16 + ln) * 64 + kk + 2*hi);
                acc[st] = wmma4(aa, bb, acc[st]);
            }
        }
        __syncthreads();
    }
#pragma unroll
    for (int st = 0; st < 4; ++st) {
#pragma unroll
        for (int r = 0; r < 8; ++r) {
            int s = st*16 + hi*8 + r;
            raw[((size_t)b * SDIM + s) * LTOT + l] = acc[st][r] + b_bcdt[s];
        }
    }
}

// -------------------------------------------------------------------------
// Kernel 2: per (b,s) plane: t = dwconv3x3(raw)+b_dw; dt = c2*t + A[s];
// softmax over 4096; AB = softmax*c0*t -> ab; t overwrites raw.
// Grid (64 s, 16 b), 256 threads.
// -------------------------------------------------------------------------
__global__ __launch_bounds__(256) void k_conv_softmax(float* __restrict__ raw,
                                                      float* __restrict__ ab,
                                                      const float* __restrict__ w_dw,
                                                      const float* __restrict__ b_dw,
                                                      const float* __restrict__ Avec,
                                                      const float* __restrict__ coeffs) {
    __shared__ float plane[LTOT];
    __shared__ float redmax[8];
    __shared__ float redsum[8];
    const int s = blockIdx.x, b = blockIdx.y, tid = threadIdx.x;
    const int lane = tid & 31, wave = tid >> 5;
    float* __restrict__ base = raw + ((size_t)b * SDIM + s) * LTOT;
    float* __restrict__ abp  = ab  + ((size_t)b * SDIM + s) * LTOT;

#pragma unroll
    for (int i = 0; i < 16; ++i) plane[tid + i*256] = base[tid + i*256];
    __syncthreads();

    float wloc[9];
#pragma unroll
    for (int i = 0; i < 9; ++i) wloc[i] = w_dw[s*9 + i];
    const float bias = b_dw[s], Aval = Avec[s];
    const float c0 = coeffs[0], c2 = coeffs[2];

    float tt[16], dt[16];
    float mx = -3.402823466e38f;
#pragma unroll
    for (int i = 0; i < 16; ++i) {
        int p = tid + i*256;
        int r = p >> 6, c = p & 63;
        float acc = bias;
#pragma unroll
        for (int dr = -1; dr <= 1; ++dr)
#pragma unroll
            for (int dc = -1; dc <= 1; ++dc) {
                int rr = r + dr, cc = c + dc;
                if (rr >= 0 && rr < 64 && cc >= 0 && cc < 64)
                    acc += wloc[(dr+1)*3 + (dc+1)] * plane[rr*64 + cc];
            }
        tt[i] = acc;
        dt[i] = c2 * acc + Aval;
        mx = fmaxf(mx, dt[i]);
    }
    // block max (wave32 shuffle tree + LDS)
#pragma unroll
    for (int off = 16; off > 0; off >>= 1) mx = fmaxf(mx, __shfl_xor(mx, off, 32));
    if (lane == 0) redmax[wave] = mx;
    __syncthreads();
    float gmx = redmax[0];
#pragma unroll
    for (int w = 1; w < 8; ++w) gmx = fmaxf(gmx, redmax[w]);

    float ee[16], sum = 0.f;
#pragma unroll
    for (int i = 0; i < 16; ++i) { ee[i] = __expf(dt[i] - gmx); sum += ee[i]; }
#pragma unroll
    for (int off = 16; off > 0; off >>= 1) sum += __shfl_xor(sum, off, 32);
    if (lane == 0) redsum[wave] = sum;
    __syncthreads();
    float gs = 0.f;
#pragma unroll
    for (int w = 0; w < 8; ++w) gs += redsum[w];
    const float inv = 1.0f / gs;

#pragma unroll
    for (int i = 0; i < 16; ++i) {
        int p = tid + i*256;
        abp[p]  = ee[i] * inv * (c0 * tt[i]);
        base[p] = tt[i];            // keep conv output for C_mat = c1*t later
    }
}

// -------------------------------------------------------------------------
// Kernel 3:  h[b,d,s] = sum_l xh[b,d,l]*AB[b,s,l]
// Grid (32 d-tiles, 16 b), 8 waves split K=4096, LDS reduce.
// -------------------------------------------------------------------------
__global__ __launch_bounds__(256) void k_h(const float* __restrict__ x,
                                           const float* __restrict__ ab,
                                           float* __restrict__ h_out) {
    __shared__ float red[8 * 4 * 256];   // 32 KB
    const int lane = threadIdx.x & 31, wave = threadIdx.x >> 5;
    const int ln = lane & 15, hi = lane >> 4;
    const int d0 = blockIdx.x * 16, b = blockIdx.y;
    const float* __restrict__ xh  = x  + (size_t)b * DM * LTOT;
    const float* __restrict__ abb = ab + (size_t)b * SDIM * LTOT;

    v8f zero = {0.f,0.f,0.f,0.f,0.f,0.f,0.f,0.f};
    v8f acc[4] = {zero, zero, zero, zero};

    const int k0 = wave * 512;
    for (int k = k0; k < k0 + 512; k += 4) {
        v2f aa = *(const v2f*)(xh + (size_t)(d0 + ln) * LTOT + k + 2*hi);
#pragma unroll
        for (int st = 0; st < 4; ++st) {
            v2f bb = *(const v2f*)(abb + (size_t)(st*16 + ln) * LTOT + k + 2*hi);
            acc[st] = wmma4(aa, bb, acc[st]);
        }
    }
#pragma unroll
    for (int st = 0; st < 4; ++st)
#pragma unroll
        for (int r = 0; r < 8; ++r)
            red[(wave*4 + st)*256 + (hi*8 + r)*16 + ln] = acc[st][r];
    __syncthreads();

    const int tid = threadIdx.x;
#pragma unroll
    for (int q = 0; q < 4; ++q) {
        int idx = tid + q*256;                 // 0..1023 outputs (16d x 64s)
        int st = idx >> 8, e = idx & 255;
        int m = e >> 4, n = e & 15;
        float sum = 0.f;
#pragma unroll
        for (int w = 0; w < 8; ++w) sum += red[(w*4 + st)*256 + e];
        h_out[((size_t)b * DM2 + d0 + m) * SDIM + st*16 + n] = sum;
    }
}

// -------------------------------------------------------------------------
// Kernel 4: fused  y_pre[d<512] = h · (c1*t) ,  y_pre[d>=512] = gate ;
//           y = dwconv3x3(y_pre, w_DW) + b_DW  — never hits HBM with y_pre.
// Grid (64 d-tiles, 8 row-chunks, 16 b); LDS tile 16ch x 10rows x 64cols.
// -------------------------------------------------------------------------
__global__ __launch_bounds__(256) void k_y_conv(const float* __restrict__ x,
                                                const float* __restrict__ h_in,
                                                const float* __restrict__ tmat,
                                                const float* __restrict__ coeffs,
                                                const float* __restrict__ w_DW,
                                                const float* __restrict__ b_DW,
                                                float* __restrict__ y_out) {
    __shared__ float tile[16 * 640];     // 40 KB: rows r0-1 .. r0+8
    const int lane = threadIdx.x & 31, wave = threadIdx.x >> 5;
    const int ln = lane & 15, hi = lane >> 4;
    const int d0 = blockIdx.x * 16, r0 = blockIdx.y * 8, b = blockIdx.z;
    const int tid = threadIdx.x;
    const int l_base = (r0 - 1) * 64;    // may be negative

    if (d0 < DM2) {
        const float c1 = coeffs[1];
        const float* __restrict__ hb = h_in + ((size_t)b * DM2 + d0) * SDIM;
        const float* __restrict__ tb = tmat + (size_t)b * SDIM * LTOT;
        v8f zero = {0.f,0.f,0.f,0.f,0.f,0.f,0.f,0.f};
        for (int nt = wave*5; nt < wave*5 + 5; ++nt) {  // 40 n-tiles of 16 l
            int l0 = l_base + nt*16;
            int row = l0 >> 6;                           // 16-chunk stays in one row
            v8f acc = zero;
            if (row >= 0 && row < 64) {
                for (int k = 0; k < SDIM; k += 4) {
                    v2f aa = *(const v2f*)(hb + (size_t)ln * SDIM + k + 2*hi);
                    aa.x *= c1; aa.y *= c1;              // fold C_mat coeff into A
                    v2f bb;
                    bb.x = tb[(size_t)(k + 2*hi)     * LTOT + l0 + ln];
                    bb.y = tb[(size_t)(k + 2*hi + 1) * LTOT + l0 + ln];
                    acc = wmma4(aa, bb, acc);
                }
            }
#pragma unroll
            for (int r = 0; r < 8; ++r)
                tile[(hi*8 + r)*640 + nt*16 + ln] = acc[r];
        }
    } else {
        // gate channels: pre-conv y is just x[:, d0.., :]
#pragma unroll
        for (int i = 0; i < 40; ++i) {
            int idx = tid + i*256;                      // 0..10239
            int m = idx / 640, ll = idx % 640;
            int lg = l_base + ll;
            float v = 0.f;
            if (lg >= 0 && lg < LTOT)
                v = x[((size_t)b * DM + d0 + m) * LTOT + lg];
            tile[m*640 + ll] = v;
        }
    }
    __syncthreads();

    // depthwise conv 3x3 from LDS, write final y
#pragma unroll
    for (int i = 0; i < 32; ++i) {
        int idx = tid + i*256;                          // 0..8191 (16ch x 8r x 64c)
        int c  = idx & 63;
        int rr = (idx >> 6) & 7;
        int m  = idx >> 9;
        int ch = d0 + m;
        float v = b_DW[ch];
#pragma unroll
        for (int dr = 0; dr < 3; ++dr) {
            int lr = rr + dr;                           // LDS rows rr..rr+2 = img r-1..r+1
#pragma unroll
            for (int dc = -1; dc <= 1; ++dc) {
                int cc = c + dc;
                if (cc >= 0 && cc < 64)
                    v += w_DW[ch*9 + dr*3 + (dc+1)] * tile[m*640 + lr*64 + cc];
            }
        }
        y_out[((size_t)b * DM + ch) * LTOT + (size_t)(r0 + rr)*64 + c] = v;
    }
}

extern "C" void kernel_launch(void* const* d_in, const int* in_sizes, int n_in,
                              void* d_out, int out_size, void* d_ws, size_t ws_size,
                              hipStream_t stream) {
    const float* x      = (const float*)d_in[0];
    const float* w_bcdt = (const float*)d_in[1];
    const float* b_bcdt = (const float*)d_in[2];
    const float* w_dw   = (const float*)d_in[3];
    const float* b_dw   = (const float*)d_in[4];
    const float* A      = (const float*)d_in[5];
    const float* coeffs = (const float*)d_in[6];
    const float* w_DW   = (const float*)d_in[7];
    const float* b_DW   = (const float*)d_in[8];
    // d_in[9], d_in[10] = H, W (hardcoded 64x64)

    float* y_out = (float*)d_out;                               // (16,1024,4096)
    float* h_out = y_out + (size_t)BATCH * DM * LTOT;           // (16,512,64)

    float* raw = (float*)d_ws;                                  // BCdt_raw -> t   (16 MiB)
    float* ab  = raw + (size_t)BATCH * SDIM * LTOT;             // AB              (16 MiB)

    k_bcdt        <<<dim3(32, 16),    256, 0, stream>>>(x, w_bcdt, b_bcdt, raw);
    k_conv_softmax<<<dim3(64, 16),    256, 0, stream>>>(raw, ab, w_dw, b_dw, A, coeffs);
    k_h           <<<dim3(32, 16),    256, 0, stream>>>(x, ab, h_out);
    k_y_conv      <<<dim3(64, 8, 16), 256, 0, stream>>>(x, h_out, raw, coeffs, w_DW, b_DW, y_out);
}